// Capsule_87522843558307
// MI455X (gfx1250) — compile-verified
//
#include <hip/hip_runtime.h>
#include <math.h>

// ---------------------------------------------------------------------------
// Capsule routing reference collapses exactly (b stays 0 through every softmax)
// to:   out[b,j,:] = squash( (1/J) * x2[b,:] @ W2[j*16+d, :]^T )
// with  x2[B=64, K=9216], W2[N=512, K=9216]  (K = i*8+m)
// Bandwidth-bound fp32 GEMM -> V_WMMA_F32_16X16X4_F32, split-K + f32 atomics.
// ---------------------------------------------------------------------------

typedef float v2f __attribute__((ext_vector_type(2)));
typedef float v8f __attribute__((ext_vector_type(8)));

#define B_DIM   64
#define I_DIM   1152
#define M_DIM   8
#define J_DIM   32
#define D_DIM   16
#define K_DIM   (I_DIM * M_DIM)     // 9216
#define N_DIM   (J_DIM * D_DIM)     // 512
#define CHUNKS  36                  // 36 * 256 = 9216
#define KCHUNK  256                 // K elems per chunk (64 wmma K-steps of 4)
#define WPB     4                   // waves per block

__global__ void caps_zero_ws(float* __restrict__ ws) {
    int i = blockIdx.x * blockDim.x + threadIdx.x;   // grid sized exactly B*N
    ws[i] = 0.0f;
}

__global__ __launch_bounds__(128) void caps_gemm_splitk(
        const float* __restrict__ x,    // [B, I, M]  == [B, K]
        const float* __restrict__ W,    // [J, I, D, M]
        float* __restrict__ ws)         // [B, N] fp32 accumulator
{
    const int lane  = threadIdx.x & 31;
    const int wave  = threadIdx.x >> 5;
    const int wid   = blockIdx.x * WPB + wave;      // 0 .. 128*CHUNKS-1 exactly
    const int chunk = wid % CHUNKS;
    const int tile  = wid / CHUNKS;                 // 0..127
    const int tile_m = tile & 3;                    // batch tile (M dim of GEMM)
    const int j      = tile >> 2;                   // output capsule (N tile)

    // A 16x4 f32 frag: lanes 0-15 hold K+{0,1}, lanes 16-31 hold K+{2,3}
    const int o    = (lane >> 4) << 1;              // 0 or 2
    const int arow = tile_m * 16 + (lane & 15);     // batch row
    const int dcol = lane & 15;                     // d within tile j

    const int kbase = chunk * KCHUNK;
    const float* pA = x + (size_t)arow * K_DIM + kbase + o;
    // W addr(j,i,d,m) = j*I*D*M + i*(D*M) + d*M + m ;  i = k>>3, m = k&7
    const float* pB = W + (size_t)j * (I_DIM * D_DIM * M_DIM)
                        + (size_t)(kbase >> 3) * (D_DIM * M_DIM)
                        + dcol * M_DIM + o;

    v8f acc = {0.f, 0.f, 0.f, 0.f, 0.f, 0.f, 0.f, 0.f};

    #pragma unroll 2
    for (int th = 0; th < KCHUNK / 8; ++th) {       // 2 K-steps per iteration
        v2f a0 = *(const v2f*)(pA);
        v2f b0 = *(const v2f*)(pB);
        v2f a1 = *(const v2f*)(pA + 4);
        v2f b1 = *(const v2f*)(pB + 4);
        __builtin_prefetch(pB + 8 * 128, 0, 1);     // stream W ~8 iters ahead
        acc = __builtin_amdgcn_wmma_f32_16x16x4_f32(
                  false, a0, false, b0, (short)0, acc, false, false);
        acc = __builtin_amdgcn_wmma_f32_16x16x4_f32(
                  false, a1, false, b1, (short)0, acc, false, false);
        pA += 8;          // +2 K-steps for x (contiguous K)
        pB += 128;        // +1 i-line for W (D*M floats)
    }

    // C/D f32 16x16 layout: VGPR g -> (M=g, N=lane) lanes 0-15 ; (M=8+g) lanes 16-31
    const int nc = j * 16 + (lane & 15);
    const int rb = tile_m * 16 + ((lane >> 4) << 3);
    float* wsp = ws + (size_t)rb * N_DIM + nc;
    #pragma unroll
    for (int g = 0; g < 8; ++g)
        unsafeAtomicAdd(wsp + (size_t)g * N_DIM, acc[g]);   // global_atomic_add_f32
}

__global__ void caps_squash(const float* __restrict__ ws, float* __restrict__ out) {
    int idx = blockIdx.x * blockDim.x + threadIdx.x;  // (b*32 + j), 0..2047 exact
    const float invJ = 1.0f / 32.0f;
    const float* p = ws + (size_t)idx * 16;
    float s[16];
    float sq = 0.f;
    #pragma unroll
    for (int d = 0; d < 16; ++d) { float v = p[d] * invJ; s[d] = v; sq += v * v; }
    float scale = sq / ((1.0f + sq) * sqrtf(sq + 1e-7f));
    float* q = out + (size_t)idx * 16;
    #pragma unroll
    for (int d = 0; d < 16; ++d) q[d] = scale * s[d];
}

extern "C" void kernel_launch(void* const* d_in, const int* in_sizes, int n_in,
                              void* d_out, int out_size, void* d_ws, size_t ws_size,
                              hipStream_t stream) {
    const float* x = (const float*)d_in[0];   // [64, 1152, 8]
    const float* W = (const float*)d_in[1];   // [32, 1152, 16, 8]
    float* ws  = (float*)d_ws;                // needs B*N*4 = 128 KB
    float* out = (float*)d_out;               // [64, 32, 16]

    caps_zero_ws   <<<(B_DIM * N_DIM) / 256, 256, 0, stream>>>(ws);
    caps_gemm_splitk<<<(128 * CHUNKS) / WPB, 128, 0, stream>>>(x, W, ws);
    caps_squash    <<<(B_DIM * J_DIM) / 256, 256, 0, stream>>>(ws, out);
}